// QuantumRefinement_65481071403327
// MI455X (gfx1250) — compile-verified
//
#include <hip/hip_runtime.h>
#include <hip/hip_bf16.h>
#include <stdint.h>

#define NQ      14
#define NL      8
#define NSTATES 16384      // 2^14 amplitudes
#define BLOCK   256        // 8 waves (wave32)

typedef __attribute__((ext_vector_type(2))) float v2f;
typedef __attribute__((ext_vector_type(8))) float v8f;

__device__ __forceinline__ float2 cmulf(float2 a, float2 b) {
  return make_float2(a.x * b.x - a.y * b.y, a.x * b.y + a.y * b.x);
}

// Lazy CNOT-chain permutation after l layers: m_l(y) = XOR_{j submask of l} (y >> j)
// (chain CNOT(q,q+1), q=0..12 == suffix-XOR on bits; inverse per layer is y ^ (y>>1),
//  and (1+s)^l over GF(2) has s^j coeff = C(l,j) mod 2 = [j submask of l]).
__device__ __forceinline__ uint32_t perm_map(uint32_t y, int l) {
  uint32_t r = y;
#pragma unroll
  for (int j = 1; j <= 7; ++j)
    if ((j & ~l) == 0) r ^= (y >> j);
  return r;
}

__global__ __launch_bounds__(BLOCK)
void qcircuit_kernel(const float* __restrict__ latent,
                     const float* __restrict__ weights,
                     float* __restrict__ out) {
  extern __shared__ float smem[];
  float2* st2  = (float2*)smem;           // 16384 complex amplitudes (128 KB)
  float* s_tab = smem + 2 * NSTATES;      // 256 floats: init tables, then reduction

  const int tid  = threadIdx.x;
  const int lane = tid & 31;
  const int wave = tid >> 5;
  const int bid  = blockIdx.x;            // one batch sample per workgroup
  const float* lat = latent + bid * NQ;

  // ---------- initial state: RX(latent) on |0..0> is a product state ----------
  // amp(y) = (-i)^popcount(y) * prod_q (bit_q ? sin(t_q/2) : cos(t_q/2))
  // split into hi-7 / lo-7 bit partial-product tables (qubit q <-> bit 13-q)
  {
    const int half = tid >> 7, idx = tid & 127;
    float prod = 1.f;
#pragma unroll
    for (int j = 0; j < 7; ++j) {
      float s, c;
      sincosf(0.5f * lat[half * 7 + j], &s, &c);
      prod *= ((idx >> (6 - j)) & 1) ? s : c;
    }
    s_tab[tid] = prod;
  }
  __syncthreads();
#pragma unroll 4
  for (int i = 0; i < NSTATES / BLOCK; ++i) {
    const int y = tid * (NSTATES / BLOCK) + i;
    const float mag = s_tab[y >> 7] * s_tab[128 + (y & 127)];
    const int pc = __popc(y) & 3;
    float2 a;
    a.x = (pc == 0) ? mag : ((pc == 2) ? -mag : 0.f);
    a.y = (pc == 1) ? -mag : ((pc == 3) ? mag : 0.f);
    st2[y] = a;
  }
  __syncthreads();

  // ---------- 8 layers: fused 4-qubit Rot unitaries via f32 WMMA ----------
  const int bsh[4] = {10, 6, 2, 0};   // low bit position of each 4-bit group
  for (int l = 0; l < NL; ++l) {
    for (int g = 0; g < 4; ++g) {
      const int b = bsh[g];
      // 2x2 complex sub-gates: k-bit i <-> global bit (b+i) <-> qubit 13-b-i.
      // Group 3 = bits {3..0} applies I (x) I (x) R12 (x) R13 (commutes with group 2).
      float2 gt[4][2][2];
#pragma unroll
      for (int i = 0; i < 4; ++i) {
        if (g == 3 && i >= 2) {
          gt[i][0][0] = make_float2(1.f, 0.f); gt[i][0][1] = make_float2(0.f, 0.f);
          gt[i][1][0] = make_float2(0.f, 0.f); gt[i][1][1] = make_float2(1.f, 0.f);
        } else {
          const int q = 13 - b - i;
          const float* w = weights + (l * NQ + q) * 3;
          const float phi = w[0], th = w[1], om = w[2];
          float sth, cth, sa, ca, sb, cb;
          sincosf(0.5f * th, &sth, &cth);
          sincosf(0.5f * (phi + om), &sa, &ca);
          sincosf(0.5f * (phi - om), &sb, &cb);
          gt[i][0][0] = make_float2( cth * ca, -cth * sa);   // e^{-i(phi+om)/2} c
          gt[i][0][1] = make_float2(-sth * cb, -sth * sb);   // -e^{ i(phi-om)/2} s
          gt[i][1][0] = make_float2( sth * cb, -sth * sb);   // e^{-i(phi-om)/2} s
          gt[i][1][1] = make_float2( cth * ca,  cth * sa);   // e^{ i(phi+om)/2} c
        }
      }
      // A fragments: fused 16x16 unitary U = (x)_i gt[i], as four 16x4 K-blocks.
      // ISA A layout: lanes 0-15 hold K=4j+{0,1}, lanes 16-31 hold K=4j+{2,3}, M=lane&15.
      const int m   = lane & 15;
      const int khi = (lane >> 4) << 1;
      v2f aR[4], aI[4], aIn[4];
#pragma unroll
      for (int j = 0; j < 4; ++j) {
        const int k0 = 4 * j + khi, k1 = k0 + 1;
        float2 u0 = gt[0][m & 1][k0 & 1];
        u0 = cmulf(u0, gt[1][(m >> 1) & 1][(k0 >> 1) & 1]);
        u0 = cmulf(u0, gt[2][(m >> 2) & 1][(k0 >> 2) & 1]);
        u0 = cmulf(u0, gt[3][(m >> 3) & 1][(k0 >> 3) & 1]);
        float2 u1 = gt[0][m & 1][k1 & 1];
        u1 = cmulf(u1, gt[1][(m >> 1) & 1][(k1 >> 1) & 1]);
        u1 = cmulf(u1, gt[2][(m >> 2) & 1][(k1 >> 2) & 1]);
        u1 = cmulf(u1, gt[3][(m >> 3) & 1][(k1 >> 3) & 1]);
        aR[j].x  =  u0.x; aR[j].y  =  u1.x;
        aI[j].x  =  u0.y; aI[j].y  =  u1.y;
        aIn[j].x = -u0.y; aIn[j].y = -u1.y;   // f32 WMMA has no A-neg: materialize -Ui
      }
      // XOR address offsets: perm is GF(2)-linear, so perm(c | k<<b) = perm(c) ^ perm(k<<b)
      const uint32_t pk1 = perm_map(1u << b, l);
      const uint32_t pk2 = perm_map(2u << b, l);
      const uint32_t pk4 = perm_map(4u << b, l);
      const uint32_t pk8 = perm_map(8u << b, l);
      const uint32_t lowm = (1u << b) - 1u;

      // 1024 columns per group = 64 tiles of 16; 8 waves x 8 tiles
      for (int ti = 0; ti < 8; ++ti) {
        const int t = wave * 8 + ti;
        const int c = t * 16 + (lane & 15);
        const uint32_t cpart = (((uint32_t)c >> b) << (b + 4)) | ((uint32_t)c & lowm);
        const uint32_t pcb = perm_map(cpart, l);
        // B fragments: 16(K) x 16(N) complex state tile gathered from LDS
        v2f bR[4], bI[4];
#pragma unroll
        for (int j = 0; j < 4; ++j) {
          const int k0 = 4 * j + khi, k1 = k0 + 1;
          uint32_t x0 = pcb, x1 = pcb;
          if (k0 & 1) x0 ^= pk1; if (k0 & 2) x0 ^= pk2;
          if (k0 & 4) x0 ^= pk4; if (k0 & 8) x0 ^= pk8;
          if (k1 & 1) x1 ^= pk1; if (k1 & 2) x1 ^= pk2;
          if (k1 & 4) x1 ^= pk4; if (k1 & 8) x1 ^= pk8;
          const float2 s0 = st2[x0], s1 = st2[x1];
          bR[j].x = s0.x; bR[j].y = s1.x;
          bI[j].x = s0.y; bI[j].y = s1.y;
        }
        // complex 16x16 matmul: Cr = Ur*Sr - Ui*Si ; Ci = Ur*Si + Ui*Sr
        v8f cr = {}, ci = {};
#pragma unroll
        for (int j = 0; j < 4; ++j)
          cr = __builtin_amdgcn_wmma_f32_16x16x4_f32(false, aR[j],  false, bR[j], (short)0, cr, false, false);
#pragma unroll
        for (int j = 0; j < 4; ++j)
          cr = __builtin_amdgcn_wmma_f32_16x16x4_f32(false, aIn[j], false, bI[j], (short)0, cr, false, false);
#pragma unroll
        for (int j = 0; j < 4; ++j)
          ci = __builtin_amdgcn_wmma_f32_16x16x4_f32(false, aR[j],  false, bI[j], (short)0, ci, false, false);
#pragma unroll
        for (int j = 0; j < 4; ++j)
          ci = __builtin_amdgcn_wmma_f32_16x16x4_f32(false, aI[j],  false, bR[j], (short)0, ci, false, false);
        // D layout: VGPR r -> M = r (lanes 0-15) / r+8 (lanes 16-31), N = lane&15
        const int mb = (lane >> 4) << 3;
#pragma unroll
        for (int r = 0; r < 8; ++r) {
          const int mm = mb + r;
          uint32_t x = pcb;
          if (mm & 1) x ^= pk1; if (mm & 2) x ^= pk2;
          if (mm & 4) x ^= pk4; if (mm & 8) x ^= pk8;
          st2[x] = make_float2(cr[r], ci[r]);
        }
      }
      __syncthreads();
    }
  }

  // ---------- <Z_q> readout (logical index y lives at phys = y ^ (y>>8)) ----------
  float acc[NQ];
#pragma unroll
  for (int q = 0; q < NQ; ++q) acc[q] = 0.f;
  for (int i = 0; i < NSTATES / BLOCK; ++i) {
    const int y = tid + BLOCK * i;
    const uint32_t phys = (uint32_t)y ^ ((uint32_t)y >> 8);   // m_8: submasks {0,8}
    const float2 a = st2[phys];
    const float p = a.x * a.x + a.y * a.y;
#pragma unroll
    for (int q = 0; q < NQ; ++q)
      acc[q] += ((y >> (13 - q)) & 1) ? -p : p;
  }
  // deterministic reduction: intra-wave shuffle tree, then 8 wave-partials in LDS
#pragma unroll
  for (int q = 0; q < NQ; ++q) {
    float v = acc[q];
#pragma unroll
    for (int off = 16; off > 0; off >>= 1) v += __shfl_down(v, off, 32);
    if (lane == 0) s_tab[wave * NQ + q] = v;
  }
  __syncthreads();
  if (tid < NQ) {
    float z = 0.f;
#pragma unroll
    for (int w = 0; w < 8; ++w) z += s_tab[w * NQ + tid];
    out[bid * NQ + tid] = z;
  }
}

extern "C" void kernel_launch(void* const* d_in, const int* in_sizes, int n_in,
                              void* d_out, int out_size, void* d_ws, size_t ws_size,
                              hipStream_t stream) {
  const float* latent  = (const float*)d_in[0];   // [512,14] f32
  const float* weights = (const float*)d_in[1];   // [8,14,3] f32
  float* out = (float*)d_out;                     // [512,14] f32

  const int smem_bytes = (2 * NSTATES + 256) * (int)sizeof(float);  // ~132 KB
  hipFuncSetAttribute(reinterpret_cast<const void*>(qcircuit_kernel),
                      hipFuncAttributeMaxDynamicSharedMemorySize, smem_bytes);
  qcircuit_kernel<<<512, BLOCK, smem_bytes, stream>>>(latent, weights, out);
}